// ConvNeXtLoss_90709709291860
// MI455X (gfx1250) — compile-verified
//
#include <hip/hip_runtime.h>

typedef __attribute__((ext_vector_type(2))) float v2f;
typedef __attribute__((ext_vector_type(4))) float v4f;
typedef __attribute__((ext_vector_type(8))) float v8f;
typedef __attribute__((ext_vector_type(4))) int   v4i;

#define BATCH 16
#define CHN   4
#define HW    (512 * 512)          // 262144 elements per (b,c) plane
#define PLANES (BATCH * CHN)       // 64
#define VEC4_PER_PLANE (HW / 4)    // 65536
#define BPP 32                     // blocks per plane
#define THREADS 256
#define VEC4_PER_BLOCK (VEC4_PER_PLANE / BPP)      // 2048
#define VEC4_PER_THREAD (VEC4_PER_BLOCK / THREADS) // 8

// workspace layout (floats)
#define WS_SPT  0     // [64] sum(p*t)   per plane
#define WS_SPP  64    // [64] sum(p*p)   per plane
#define WS_ST   128   // [64] sum(t)     per plane
#define WS_SP   192   // [64] sum(p)     per plane
#define WS_SPOS 256   // [16] BCE t==1 unweighted contribution sum per batch
#define WS_SALL 272   // [16] BCE total unweighted contribution sum per batch
#define WS_TOTAL 288

#define EPSF    1e-14f
#define SMOOTHF 1e-6f
#define LN2F    0.69314718055994530942f

// Full-precision fp32 wave32 reduction via V_WMMA_F32_16X16X4_F32.
// A (16x4 f32, 2 VGPR): lanes 0-15 supply (M=lane,K=0), lanes 16-31 (M=lane-16,K=2).
// With A.x = val, A.y = 0, B = ones: D[m][n] = val[m] + val[m+16].
// Per-lane sum of the 8 D VGPRs -> lane 0 holds sum(rows 0-7), lane 16 sum(rows 8-15);
// their sum is the full 32-lane reduction.
__device__ __forceinline__ float wave_reduce_wmma(float v) {
    v2f a; a.x = v;    a.y = 0.0f;
    v2f b; b.x = 1.0f; b.y = 1.0f;
    v8f c = {};
    v8f d = __builtin_amdgcn_wmma_f32_16x16x4_f32(
        /*neg_a=*/false, a, /*neg_b=*/false, b,
        /*c_mod=*/(short)0, c, /*reuse_a=*/false, /*reuse_b=*/false);
    return d[0] + d[1] + d[2] + d[3] + d[4] + d[5] + d[6] + d[7];
}

__global__ void init_ws_kernel(float* ws) {
    int i = blockIdx.x * blockDim.x + threadIdx.x;
    if (i < WS_TOTAL) ws[i] = 0.0f;
}

__global__ __launch_bounds__(THREADS) void loss_main_kernel(
    const float* __restrict__ cls, const int* __restrict__ lab, float* __restrict__ ws) {
    const int plane = blockIdx.x / BPP;          // 0..63  (b*4 + c)
    const int chunk = blockIdx.x % BPP;
    const int batch = plane >> 2;
    const size_t base =
        (size_t)plane * VEC4_PER_PLANE + (size_t)chunk * VEC4_PER_BLOCK + threadIdx.x;

    const v4f* __restrict__ p4 = (const v4f*)cls;
    const v4i* __restrict__ t4 = (const v4i*)lab;

    float s_pt = 0.0f, s_pp = 0.0f, s_t = 0.0f, s_p = 0.0f;
    float s_pos = 0.0f, s_all = 0.0f;   // BCE contributions: t==1 only, and all

#pragma unroll
    for (int j = 0; j < VEC4_PER_THREAD; ++j) {
        const size_t idx = base + (size_t)j * THREADS;
        // streamed once: non-temporal hint keeps L2 clean (TH=NT)
        v4f pv = __builtin_nontemporal_load(&p4[idx]);
        v4i tv = __builtin_nontemporal_load(&t4[idx]);
#pragma unroll
        for (int e = 0; e < 4; ++e) {
            const float p   = pv[e];
            const int   ti  = tv[e];            // label in {0,1}
            const float tf  = (float)ti;        // one v_cvt_f32_i32
            const bool  tt  = (ti != 0);
            const float omp = 1.0f - p;

            s_pp = __builtin_fmaf(p, p, s_pp);
            s_pt = __builtin_fmaf(p, tf, s_pt);
            s_p += p;
            s_t += tf;

            // BCE live branch: q = log arg, r = weight-exponent arg (same pair, swapped)
            const float q  = tt ? p : omp;
            const float r  = tt ? omp : p;
            const float rc = fmaxf(r, EPSF);    // upper clip 1-1e-14 == 1.0f in fp32
            // weight 8^(rc^0.5) = exp2(3*sqrt(rc)); raw HW trans ops, args in-range
            const float w  = __builtin_amdgcn_exp2f(3.0f * __builtin_amdgcn_sqrtf(rc));
            // -max(log(q),-100) == min(-ln2*log2(q), 100); log2(0)=-inf -> 100
            const float bce = fminf(-LN2F * __builtin_amdgcn_logf(q), 100.0f);
            const float contrib = w * bce;

            s_all += contrib;
            s_pos = __builtin_fmaf(tf, contrib, s_pos);
        }
    }

    // ---- block reduction: per-wave WMMA reduce, lanes 0 & 16 hold partials ----
    __shared__ float red[6][16];
    const int wave = threadIdx.x >> 5;
    const int lane = threadIdx.x & 31;
    float acc[6] = {s_pt, s_pp, s_t, s_p, s_pos, s_all};
#pragma unroll
    for (int k = 0; k < 6; ++k) {
        float part = wave_reduce_wmma(acc[k]);   // EXEC all-ones here
        if ((lane & 15) == 0) red[k][wave * 2 + (lane >> 4)] = part;
    }
    __syncthreads();

    if (threadIdx.x < 6) {
        const int k = threadIdx.x;
        float s = 0.0f;
#pragma unroll
        for (int i = 0; i < 16; ++i) s += red[k][i];
        if (k < 4) {
            atomicAdd(&ws[k * 64 + plane], s);
        } else if (k == 4) {
            atomicAdd(&ws[WS_SPOS + batch], s);
        } else {
            atomicAdd(&ws[WS_SALL + batch], s);
        }
    }
}

__global__ void loss_finalize_kernel(const float* __restrict__ ws, float* __restrict__ out) {
    __shared__ float vals[PLANES];
    const int j = threadIdx.x;   // 64 threads, one per plane
    const float spt = ws[WS_SPT + j];
    const float spp = ws[WS_SPP + j];
    const float st  = ws[WS_ST + j];
    const float sp  = ws[WS_SP + j];
    const float N   = (float)HW;

    // dice(p, t):  1 - (2*S_pt + sm) / (S_pp + S_t + sm)      (t^2 == t)
    const float dice = 1.0f - (2.0f * spt + SMOOTHF) / (spp + st + SMOOTHF);
    // dice(1-p, 1-t): inter' = N - S_p - S_t + S_pt ; denom' = 2N - 2S_p + S_pp - S_t
    const float ip = N - sp - st + spt;
    const float dp = 2.0f * N - 2.0f * sp + spp - st;
    const float rdice = 1.0f - (2.0f * ip + SMOOTHF) / (dp + SMOOTHF);

    float v = 2500.0f * (dice + rdice);   // DICE_COEFF == REV_DICE_COEFF == 2500

    if (j < BATCH) {
        const float npos = ws[WS_ST + 4 * j + 0] + ws[WS_ST + 4 * j + 1] +
                           ws[WS_ST + 4 * j + 2] + ws[WS_ST + 4 * j + 3];
        const float tot   = (float)(CHN * HW);
        const float alpha = (tot - npos) / tot;
        const float sxp   = ws[WS_SPOS + j];
        const float sy    = ws[WS_SALL + j] - sxp;
        v += alpha * sxp + (1.0f - alpha) * sy;    // ATTENTION_COEFF = 1
    }
    vals[j] = v;
    __syncthreads();
    if (j == 0) {
        float s = 0.0f;
        for (int k = 0; k < PLANES; ++k) s += vals[k];
        out[0] = s;
    }
}

extern "C" void kernel_launch(void* const* d_in, const int* in_sizes, int n_in,
                              void* d_out, int out_size, void* d_ws, size_t ws_size,
                              hipStream_t stream) {
    (void)in_sizes; (void)n_in; (void)out_size; (void)ws_size;
    const float* cls = (const float*)d_in[0];
    const int*   lab = (const int*)d_in[1];
    float* out = (float*)d_out;
    float* ws  = (float*)d_ws;

    init_ws_kernel<<<(WS_TOTAL + 255) / 256, 256, 0, stream>>>(ws);
    loss_main_kernel<<<PLANES * BPP, THREADS, 0, stream>>>(cls, lab, ws);
    loss_finalize_kernel<<<1, PLANES, 0, stream>>>(ws, out);
}